// GINE_NoEmbedding_19473381720871
// MI455X (gfx1250) — compile-verified
//
#include <hip/hip_runtime.h>

#define NN 50000
#define EE 400000
#define GG 512
#define FNODE 128
#define EMB 256
#define TT 10
#define LL 4

typedef float v2f __attribute__((ext_vector_type(2)));
typedef float v8f __attribute__((ext_vector_type(8)));

// ---------------- utility kernels ----------------

__global__ __launch_bounds__(256) void k_zero(float* p, long n) {
    long i = (long)blockIdx.x * 256 + threadIdx.x;
    if (i < n) p[i] = 0.0f;
}

__global__ __launch_bounds__(256) void k_count(const int* __restrict__ idx, int* __restrict__ cnt, int n) {
    int i = blockIdx.x * 256 + threadIdx.x;
    if (i < n) atomicAdd(&cnt[idx[i]], 1);
}

// ---------------- WMMA f32 GEMM: C[M,Nc] = act(A[M,K] @ W[K,Nc] + bias) ----------------
// One wave computes a 32x64 output tile (2 A fragments x 4 B fragments = 8 accumulators,
// so each B fragment is reused twice: 8 wmma per 12 lane-loads in the k-loop).
// Nc % 64 == 0, K % 4 == 0. Row tail (M % 32) handled by clamped loads + guarded stores.
// A fragment (16x4 f32): lane t (tl=t&15, th=t>>4) holds A[m=tl][k0+2*th+j], j=0,1
// B fragment (4x16 f32): lane t holds B[k0+2*th+j][n=tl]
// C fragment (16x16 f32, v8f): element r of lane t is C[m=r+8*th][n=tl]
__global__ __launch_bounds__(32) void k_gemm_wmma(
    const float* __restrict__ A, const float* __restrict__ W,
    const float* __restrict__ bias, float* __restrict__ C,
    int M, int K, int Nc, int relu)
{
    const int t  = threadIdx.x;
    const int tl = t & 15;
    const int th = t >> 4;
    const long row0 = (long)blockIdx.x * 32;
    const int  col0 = blockIdx.y * 64;

    long ra = row0 + tl;
    long rb = row0 + 16 + tl;
    const float* arow0 = A + (ra < M ? ra : (long)(M - 1)) * (long)K;
    const float* arow1 = A + (rb < M ? rb : (long)(M - 1)) * (long)K;

    v8f acc[8];
#pragma unroll
    for (int i = 0; i < 8; ++i) acc[i] = v8f{};

    for (int k0 = 0; k0 < K; k0 += 4) {
        const int ka = k0 + 2 * th;
        v2f a0, a1;
        a0.x = arow0[ka]; a0.y = arow0[ka + 1];
        a1.x = arow1[ka]; a1.y = arow1[ka + 1];
        const float* wp = W + (long)ka * Nc + col0 + tl;
        v2f b0, b1, b2, b3;
        b0.x = wp[0];   b0.y = wp[Nc + 0];
        b1.x = wp[16];  b1.y = wp[Nc + 16];
        b2.x = wp[32];  b2.y = wp[Nc + 32];
        b3.x = wp[48];  b3.y = wp[Nc + 48];
        acc[0] = __builtin_amdgcn_wmma_f32_16x16x4_f32(false, a0, false, b0, (short)0, acc[0], false, false);
        acc[1] = __builtin_amdgcn_wmma_f32_16x16x4_f32(false, a0, false, b1, (short)0, acc[1], false, false);
        acc[2] = __builtin_amdgcn_wmma_f32_16x16x4_f32(false, a0, false, b2, (short)0, acc[2], false, false);
        acc[3] = __builtin_amdgcn_wmma_f32_16x16x4_f32(false, a0, false, b3, (short)0, acc[3], false, false);
        acc[4] = __builtin_amdgcn_wmma_f32_16x16x4_f32(false, a1, false, b0, (short)0, acc[4], false, false);
        acc[5] = __builtin_amdgcn_wmma_f32_16x16x4_f32(false, a1, false, b1, (short)0, acc[5], false, false);
        acc[6] = __builtin_amdgcn_wmma_f32_16x16x4_f32(false, a1, false, b2, (short)0, acc[6], false, false);
        acc[7] = __builtin_amdgcn_wmma_f32_16x16x4_f32(false, a1, false, b3, (short)0, acc[7], false, false);
    }

#pragma unroll
    for (int half = 0; half < 2; ++half) {
#pragma unroll
        for (int c = 0; c < 4; ++c) {
            const int col = col0 + 16 * c + tl;
            const float bv = bias ? bias[col] : 0.0f;
#pragma unroll
            for (int r = 0; r < 8; ++r) {
                const long row = row0 + 16 * half + 8 * th + r;
                if (row < M) {
                    float v = acc[4 * half + c][r] + bv;
                    if (relu) v = fmaxf(v, 0.0f);
                    C[row * (long)Nc + col] = v;
                }
            }
        }
    }
}

// ---------------- BatchNorm (training-mode batch stats) ----------------

__global__ __launch_bounds__(256) void k_colstats(
    const float* __restrict__ T, int M, int ldc,
    const float* __restrict__ g, const float* __restrict__ be,
    float* __restrict__ scale, float* __restrict__ shift)
{
    const int col = blockIdx.x;
    const int tid = threadIdx.x;
    float s = 0.0f, s2 = 0.0f;
    for (long i = tid; i < M; i += 256) {
        float v = T[i * (long)ldc + col];
        s += v; s2 += v * v;
    }
    __shared__ float sh[256], sh2[256];
    sh[tid] = s; sh2[tid] = s2;
    __syncthreads();
    for (int o = 128; o > 0; o >>= 1) {
        if (tid < o) { sh[tid] += sh[tid + o]; sh2[tid] += sh2[tid + o]; }
        __syncthreads();
    }
    if (tid == 0) {
        float m = sh[0] / (float)M;
        float v = sh2[0] / (float)M - m * m;
        float sc = g[col] * rsqrtf(v + 1e-5f);
        scale[col] = sc;
        shift[col] = be[col] - m * sc;
    }
}

__global__ __launch_bounds__(256) void k_bnrelu(
    float* __restrict__ T, const float* __restrict__ scale,
    const float* __restrict__ shift, long n, int mask)
{
    long idx = (long)blockIdx.x * 256 + threadIdx.x;
    if (idx >= n) return;
    int col = (int)(idx & mask);
    T[idx] = fmaxf(T[idx] * scale[col] + shift[col], 0.0f);
}

// ---------------- edge path ----------------

// e_loop[j] = relu(4*W1[0][:] + b1) @ W2[:,j] + b2[j]   (self-loop attr is (4,0))
__global__ __launch_bounds__(256) void k_eloop(
    const float* __restrict__ W1, const float* __restrict__ b1,
    const float* __restrict__ W2, const float* __restrict__ b2,
    float* __restrict__ eloop)
{
    __shared__ float hk[512];
    const int tid = threadIdx.x;
    for (int k = tid; k < 512; k += 256)
        hk[k] = fmaxf(4.0f * W1[k] + b1[k], 0.0f);
    __syncthreads();
    float s = b2[tid];
    for (int k = 0; k < 512; ++k)
        s += hk[k] * W2[k * 256 + tid];
    eloop[tid] = s;
}

// scatter relu(ea@W1 + b1) hiddens into S[N,512] (linearity trick: W2 GEMM after scatter)
__global__ __launch_bounds__(256) void k_edge_scatter(
    const float* __restrict__ ea, const int* __restrict__ dst,
    const float* __restrict__ W1, const float* __restrict__ b1,
    float* __restrict__ S)
{
    long idx = (long)blockIdx.x * 256 + threadIdx.x;
    if (idx >= (long)EE * 512) return;
    int e = (int)(idx >> 9);
    int j = (int)(idx & 511);
    float a = ea[2 * e], b = ea[2 * e + 1];
    float h = fmaxf(a * W1[j] + b * W1[512 + j] + b1[j], 0.0f);
    atomicAdd(&S[(long)dst[e] * 512 + j], h);
}

// Hs[dst] += h[src]
__global__ __launch_bounds__(256) void k_hs_scatter(
    const int* __restrict__ src, const int* __restrict__ dst,
    const float* __restrict__ h, float* __restrict__ Hs)
{
    long idx = (long)blockIdx.x * 256 + threadIdx.x;
    if (idx >= (long)EE * 256) return;
    int e = (int)(idx >> 8);
    int j = (int)(idx & 255);
    atomicAdd(&Hs[(long)dst[e] * 256 + j], h[(long)src[e] * 256 + j]);
}

// aggr (in-place on T): T += h (self loop) + Hs + e_loop + indeg*b2
__global__ __launch_bounds__(256) void k_combine(
    float* __restrict__ T, const float* __restrict__ h,
    const float* __restrict__ Hs, const float* __restrict__ eloop,
    const float* __restrict__ b2, const int* __restrict__ indeg)
{
    long idx = (long)blockIdx.x * 256 + threadIdx.x;
    if (idx >= (long)NN * 256) return;
    int i = (int)(idx >> 8);
    int j = (int)(idx & 255);
    T[idx] += h[idx] + Hs[idx] + eloop[j] + (float)indeg[i] * b2[j];
}

// ---------------- pooling / readout ----------------

__global__ __launch_bounds__(256) void k_pool(
    const float* __restrict__ h, const int* __restrict__ batch, float* __restrict__ pooled)
{
    long idx = (long)blockIdx.x * 256 + threadIdx.x;
    if (idx >= (long)NN * 256) return;
    int i = (int)(idx >> 8);
    int j = (int)(idx & 255);
    atomicAdd(&pooled[(long)batch[i] * 256 + j], h[idx]);
}

// out[g,t] += pooled[g,:] @ lw[:,t] + cnt[g]*lb[t]   (bias counted once per node)
__global__ __launch_bounds__(256) void k_readout(
    const float* __restrict__ pooled, const int* __restrict__ cnt,
    const float* __restrict__ lw, const float* __restrict__ lb, float* __restrict__ out)
{
    int idx = blockIdx.x * 256 + threadIdx.x;
    if (idx >= GG * TT) return;
    int g = idx / TT, t = idx % TT;
    float s = (float)cnt[g] * lb[t];
    const float* pg = pooled + (long)g * 256;
    for (int j = 0; j < 256; ++j)
        s += pg[j] * lw[j * TT + t];
    out[idx] += s;
}

// ---------------- host launch ----------------

static inline unsigned gblk(long n) { return (unsigned)((n + 255) / 256); }

extern "C" void kernel_launch(void* const* d_in, const int* in_sizes, int n_in,
                              void* d_out, int out_size, void* d_ws, size_t ws_size,
                              hipStream_t stream) {
    const float* x         = (const float*)d_in[0];
    const float* edge_attr = (const float*)d_in[1];
    const int*   eidx      = (const int*)d_in[2];
    const int*   src       = eidx;
    const int*   dst       = eidx + EE;
    const int*   batch     = (const int*)d_in[3];
    const float* fh_w1 = (const float*)d_in[4];
    const float* fh_b1 = (const float*)d_in[5];
    const float* fh_g1 = (const float*)d_in[6];
    const float* fh_be1= (const float*)d_in[7];
    const float* fh_w2 = (const float*)d_in[8];
    const float* fh_b2 = (const float*)d_in[9];
    const float* fh_g2 = (const float*)d_in[10];
    const float* fh_be2= (const float*)d_in[11];
    const float* edge_w1 = (const float*)d_in[12];
    const float* edge_b1 = (const float*)d_in[13];
    const float* edge_w2 = (const float*)d_in[14];
    const float* edge_b2 = (const float*)d_in[15];
    const float* mlp_w1  = (const float*)d_in[16];
    const float* mlp_b1  = (const float*)d_in[17];
    const float* mlp_w2  = (const float*)d_in[18];
    const float* mlp_b2  = (const float*)d_in[19];
    const float* lin_w   = (const float*)d_in[20];
    const float* lin_b   = (const float*)d_in[21];
    float* out = (float*)d_out;

    // workspace layout (floats)
    float* p = (float*)d_ws;
    float* buf512 = p; p += (size_t)NN * 512;   // t1 / S / u  (shared, sequential use)
    float* hA     = p; p += (size_t)NN * 256;
    float* hB     = p; p += (size_t)NN * 256;
    float* Hs     = p; p += (size_t)NN * 256;
    float* Tg     = p; p += (size_t)NN * 256;   // S@W2, then aggr in place
    float* pooled = p; p += (size_t)GG * 256;
    float* scale  = p; p += 512;
    float* shift  = p; p += 512;
    float* eloop  = p; p += 256;
    int*   indeg  = (int*)p; p += NN;
    int*   cnt    = (int*)p; p += GG;

    const unsigned rowblk = (NN + 31) / 32;

    // init (every call: workspace/out state must not leak across replays)
    k_zero<<<gblk(GG * TT), 256, 0, stream>>>(out, GG * TT);
    k_zero<<<gblk(NN), 256, 0, stream>>>((float*)indeg, NN);
    k_zero<<<gblk(GG), 256, 0, stream>>>((float*)cnt, GG);
    k_count<<<gblk(EE), 256, 0, stream>>>(dst, indeg, EE);
    k_count<<<gblk(NN), 256, 0, stream>>>(batch, cnt, NN);

    // ---- first_h: Linear-BN-ReLU-Linear-BN-ReLU ----
    k_gemm_wmma<<<dim3(rowblk, 512 / 64), 32, 0, stream>>>(x, fh_w1, fh_b1, buf512, NN, FNODE, 512, 0);
    k_colstats<<<512, 256, 0, stream>>>(buf512, NN, 512, fh_g1, fh_be1, scale, shift);
    k_bnrelu<<<gblk((long)NN * 512), 256, 0, stream>>>(buf512, scale, shift, (long)NN * 512, 511);
    k_gemm_wmma<<<dim3(rowblk, 256 / 64), 32, 0, stream>>>(buf512, fh_w2, fh_b2, hA, NN, 512, 256, 0);
    k_colstats<<<256, 256, 0, stream>>>(hA, NN, 256, fh_g2, fh_be2, scale, shift);
    k_bnrelu<<<gblk((long)NN * 256), 256, 0, stream>>>(hA, scale, shift, (long)NN * 256, 255);

    // ---- layer-0 readout ----
    k_zero<<<gblk((long)GG * 256), 256, 0, stream>>>(pooled, (long)GG * 256);
    k_pool<<<gblk((long)NN * 256), 256, 0, stream>>>(hA, batch, pooled);
    k_readout<<<gblk(GG * TT), 256, 0, stream>>>(pooled, cnt, lin_w, lin_b, out);

    // ---- 4 GINE conv layers ----
    float* hc = hA;
    float* hn = hB;
    for (int l = 0; l < LL; ++l) {
        const float* ew1 = edge_w1 + (size_t)l * 2 * 512;
        const float* eb1 = edge_b1 + (size_t)l * 512;
        const float* ew2 = edge_w2 + (size_t)l * 512 * 256;
        const float* eb2 = edge_b2 + (size_t)l * 256;
        const float* mw1 = mlp_w1 + (size_t)l * 256 * 512;
        const float* mb1 = mlp_b1 + (size_t)l * 512;
        const float* mw2 = mlp_w2 + (size_t)l * 512 * 256;
        const float* mb2 = mlp_b2 + (size_t)l * 256;

        k_zero<<<gblk((long)NN * 512), 256, 0, stream>>>(buf512, (long)NN * 512);
        k_zero<<<gblk((long)NN * 256), 256, 0, stream>>>(Hs, (long)NN * 256);
        k_eloop<<<1, 256, 0, stream>>>(ew1, eb1, ew2, eb2, eloop);
        // scatter relu hiddens (linearity: defer @W2 until after aggregation)
        k_edge_scatter<<<gblk((long)EE * 512), 256, 0, stream>>>(edge_attr, dst, ew1, eb1, buf512);
        k_hs_scatter<<<gblk((long)EE * 256), 256, 0, stream>>>(src, dst, hc, Hs);
        // T = S @ W2  (b2 folded in via indeg term)
        k_gemm_wmma<<<dim3(rowblk, 256 / 64), 32, 0, stream>>>(buf512, ew2, nullptr, Tg, NN, 512, 256, 0);
        k_combine<<<gblk((long)NN * 256), 256, 0, stream>>>(Tg, hc, Hs, eloop, eb2, indeg);
        // node MLP
        k_gemm_wmma<<<dim3(rowblk, 512 / 64), 32, 0, stream>>>(Tg, mw1, mb1, buf512, NN, 256, 512, 1);
        k_gemm_wmma<<<dim3(rowblk, 256 / 64), 32, 0, stream>>>(buf512, mw2, mb2, hn, NN, 512, 256, 0);
        // pooled readout
        k_zero<<<gblk((long)GG * 256), 256, 0, stream>>>(pooled, (long)GG * 256);
        k_pool<<<gblk((long)NN * 256), 256, 0, stream>>>(hn, batch, pooled);
        k_readout<<<gblk(GG * TT), 256, 0, stream>>>(pooled, cnt,
                                                     lin_w + (size_t)(l + 1) * 256 * TT,
                                                     lin_b + (size_t)(l + 1) * TT, out);
        float* tmp = hc; hc = hn; hn = tmp;
    }
}